// TopicDecoder_Pt_25151328485515
// MI455X (gfx1250) — compile-verified
//
#include <hip/hip_runtime.h>
#include <math.h>

typedef __attribute__((ext_vector_type(2))) float v2f;
typedef __attribute__((ext_vector_type(8))) float v8f;
typedef __attribute__((ext_vector_type(4))) int   v4i;

#define BB   32
#define TUU  128
#define TZZ  64
#define TT   192          // TUU + TZZ
#define VV   32000
#define HH   512
#define EE   512
#define SS   1024         // HH + EE
#define NCLS 32192        // VV + TUU + TZZ

// workspace layout (float offsets)
#define WS_HW1    0                          // BB*HH
#define WS_SCORES (WS_HW1 + BB*HH)           // BB*TT
#define WS_X      (WS_SCORES + BB*TT)        // BB*SS
#define WS_HNEW   (WS_X + BB*SS)             // BB*HH
#define WS_ST     (WS_HNEW + BB*HH)          // BB*SS
#define WS_SF     (WS_ST + BB*SS)            // BB*NCLS
#define WS_MZ     (WS_SF + BB*NCLS)          // BB*2

// ---------------- async global->LDS (CDNA5), with safe fallback -------------
#if __has_builtin(__builtin_amdgcn_global_load_async_to_lds_b128)
#define ASYNC_LDS 1
#else
#define ASYNC_LDS 0
#endif

__device__ __forceinline__ void ld_async16(float* lds, const float* g) {
#if ASYNC_LDS
  __builtin_amdgcn_global_load_async_to_lds_b128(
      (__attribute__((address_space(1))) v4i*)g,
      (__attribute__((address_space(3))) v4i*)lds,
      0, 0);
#else
  *(float4*)lds = *(const float4*)g;
#endif
}

__device__ __forceinline__ void async_wait0() {
#if ASYNC_LDS
#if __has_builtin(__builtin_amdgcn_s_wait_asynccnt)
  __builtin_amdgcn_s_wait_asynccnt(0);
#else
  asm volatile("s_wait_asynccnt 0x0" ::: "memory");
#endif
#endif
}

// wait until at most 5 async ops outstanding (the next-chunk prefetch ops)
__device__ __forceinline__ void async_wait5() {
#if ASYNC_LDS
#if __has_builtin(__builtin_amdgcn_s_wait_asynccnt)
  __builtin_amdgcn_s_wait_asynccnt(5);
#else
  asm volatile("s_wait_asynccnt 0x5" ::: "memory");
#endif
#endif
}

// -------------------- K1: hW1[b,i] = W_attn[i,0:H] . h_b + b_attn[i] ---------
__global__ __launch_bounds__(256) void k_hw1(const float* __restrict__ Wattn,
    const float* __restrict__ battn, const float* __restrict__ lasth,
    float* __restrict__ hW1)
{
  __shared__ float sh[HH];
  int b = blockIdx.x;
  for (int i = threadIdx.x; i < HH; i += 256) sh[i] = lasth[b*HH + i];
  __syncthreads();
  for (int i = threadIdx.x; i < HH; i += 256) {
    const float* w = Wattn + (size_t)i * (2*HH);
    float acc = battn[i];
    #pragma unroll 4
    for (int j = 0; j < HH; ++j) acc += w[j] * sh[j];
    hW1[b*HH + i] = acc;
  }
}

// ---- K2: scores[b,t] = sum_i tanh(hW1[b,i] + enc[b,t,:].W2[i,:]) * v_attn[i]
// WMMA fp32: M=16 rows (t-tile), N=512 (i), K=512 (enc dim). 8 waves x 64 cols.
__global__ __launch_bounds__(256) void k_attn_scores(
    const float* __restrict__ uh, const float* __restrict__ pzh,
    const float* __restrict__ Wattn, const float* __restrict__ hW1,
    const float* __restrict__ vattn, float* __restrict__ scores)
{
  __shared__ __align__(16) float sA[16][HH + 4];   // row stride 516*4B (16B mult)
  __shared__ float sScore[16];
  int b  = blockIdx.x / (TT/16);
  int t0 = (blockIdx.x % (TT/16)) * 16;

  // async-stage the 16x512 encoder tile (2048 float4 units, 8 per thread)
  {
    int unit = threadIdx.x;
    #pragma unroll
    for (int u = 0; u < 8; ++u) {
      int r = unit >> 7, kq = unit & 127;
      int t = t0 + r;
      const float* src = (t < TUU)
          ? (uh  + ((size_t)b*TUU + t)*HH        + kq*4)
          : (pzh + ((size_t)b*TZZ + (t - TUU))*HH + kq*4);
      ld_async16(&sA[r][kq*4], src);
      unit += 256;
    }
  }
  if (threadIdx.x < 16) sScore[threadIdx.x] = 0.0f;
  async_wait0();
  __syncthreads();

  int lane = threadIdx.x & 31;
  int wave = threadIdx.x >> 5;
  int half = lane >> 4, lid = lane & 15;
  int n0w  = wave * 64;

  v8f zacc = {};
  v8f acc[4] = {zacc, zacc, zacc, zacc};

  for (int k0 = 0; k0 < HH; k0 += 4) {
    v2f a;
    a.x = sA[lid][k0 + 2*half];
    a.y = sA[lid][k0 + 2*half + 1];
    #pragma unroll
    for (int ct = 0; ct < 4; ++ct) {
      int n = n0w + ct*16 + lid;
      const float* bp = Wattn + (size_t)n*(2*HH) + HH + k0 + 2*half;
      v2f bf; bf.x = bp[0]; bf.y = bp[1];
      acc[ct] = __builtin_amdgcn_wmma_f32_16x16x4_f32(false, a, false, bf,
                  (short)0, acc[ct], false, false);
    }
  }

  float p[8];
  #pragma unroll
  for (int r = 0; r < 8; ++r) p[r] = 0.0f;
  #pragma unroll
  for (int ct = 0; ct < 4; ++ct) {
    int n = n0w + ct*16 + lid;
    float hb = hW1[b*HH + n];
    float va = vattn[n];
    #pragma unroll
    for (int r = 0; r < 8; ++r)
      p[r] += tanhf(acc[ct][r] + hb) * va;
  }
  #pragma unroll
  for (int r = 0; r < 8; ++r)
    for (int off = 8; off >= 1; off >>= 1)
      p[r] += __shfl_xor(p[r], off, 32);
  if (lid == 0) {
    #pragma unroll
    for (int r = 0; r < 8; ++r)
      atomicAdd(&sScore[r + 8*half], p[r]);
  }
  __syncthreads();
  if (threadIdx.x < 16)
    scores[b*TT + t0 + threadIdx.x] = sScore[threadIdx.x];
}

// ---------------- K3: softmax over t, context, x = [emb, context] -----------
__global__ __launch_bounds__(256) void k_softmax_ctx(
    const float* __restrict__ uh, const float* __restrict__ pzh,
    const float* __restrict__ emb, const float* __restrict__ scores,
    float* __restrict__ x)
{
  __shared__ float sat[TT];
  __shared__ float red[256];
  int b = blockIdx.x, tid = threadIdx.x;
  float v = (tid < TT) ? scores[b*TT + tid] : -3.0e38f;
  red[tid] = v;
  __syncthreads();
  for (int s = 128; s > 0; s >>= 1) {
    if (tid < s) red[tid] = fmaxf(red[tid], red[tid + s]);
    __syncthreads();
  }
  float mx = red[0];
  __syncthreads();
  float e = (tid < TT) ? expf(v - mx) : 0.0f;
  red[tid] = e;
  __syncthreads();
  for (int s = 128; s > 0; s >>= 1) {
    if (tid < s) red[tid] += red[tid + s];
    __syncthreads();
  }
  float invZ = 1.0f / red[0];
  if (tid < TT) sat[tid] = e * invZ;
  __syncthreads();
  for (int h = tid; h < HH; h += 256) {
    float c = 0.0f;
    for (int t = 0; t < TT; ++t) {
      float ev = (t < TUU) ? uh[((size_t)b*TUU + t)*HH + h]
                           : pzh[((size_t)b*TZZ + (t - TUU))*HH + h];
      c += sat[t] * ev;
    }
    x[b*SS + EE + h] = c;
  }
  for (int e2 = tid; e2 < EE; e2 += 256)
    x[b*SS + e2] = emb[b*EE + e2];
}

// ------------------------------- K4: GRU cell -------------------------------
__global__ __launch_bounds__(256) void k_gru(
    const float* __restrict__ Wih, const float* __restrict__ Whh,
    const float* __restrict__ bih, const float* __restrict__ bhh,
    const float* __restrict__ x, const float* __restrict__ lasth,
    const float* __restrict__ emb,
    float* __restrict__ st, float* __restrict__ hnew_ws,
    float* __restrict__ out_nlh, float* __restrict__ out_gru)
{
  __shared__ float sx[SS];
  __shared__ float shp[HH];
  int b = blockIdx.x, tid = threadIdx.x;
  for (int i = tid; i < SS; i += 256) sx[i] = x[b*SS + i];
  for (int i = tid; i < HH; i += 256) shp[i] = lasth[b*HH + i];
  __syncthreads();
  for (int j = tid; j < HH; j += 256) {
    float gi_r = bih[j], gi_z = bih[HH + j], gi_n = bih[2*HH + j];
    const float* wr = Wih + (size_t)j*SS;
    const float* wz = Wih + (size_t)(HH + j)*SS;
    const float* wn = Wih + (size_t)(2*HH + j)*SS;
    #pragma unroll 4
    for (int k = 0; k < SS; ++k) {
      float xv = sx[k];
      gi_r += wr[k]*xv; gi_z += wz[k]*xv; gi_n += wn[k]*xv;
    }
    float gh_r = bhh[j], gh_z = bhh[HH + j], gh_n = bhh[2*HH + j];
    const float* vr = Whh + (size_t)j*HH;
    const float* vz = Whh + (size_t)(HH + j)*HH;
    const float* vn = Whh + (size_t)(2*HH + j)*HH;
    #pragma unroll 4
    for (int k = 0; k < HH; ++k) {
      float hv = shp[k];
      gh_r += vr[k]*hv; gh_z += vz[k]*hv; gh_n += vn[k]*hv;
    }
    float r = 1.0f/(1.0f + expf(-(gi_r + gh_r)));
    float z = 1.0f/(1.0f + expf(-(gi_z + gh_z)));
    float n = tanhf(gi_n + r*gh_n);
    float hv = shp[j];
    float hn = (1.0f - z)*n + z*hv;
    hnew_ws[b*HH + j] = hn;
    st[b*SS + j]      = hn;
    st[b*SS + HH + j] = emb[b*EE + j];
    out_nlh[b*HH + j] = hn;
    out_gru[b*HH + j] = hn;
  }
}

// --------- K5: score_g = st @ W_gen^T + b_gen (32 x 32000 x 1024) -----------
// Double-buffered async pipeline: prefetch chunk i+1 while computing chunk i.
// Each thread issues exactly 5 async B128 ops per chunk (1 sSt + 4 sW), so
// s_wait_asynccnt<=5 guarantees the current chunk has landed in LDS.
#define GEN_KC 32
__global__ __launch_bounds__(256) void k_gen(
    const float* __restrict__ st, const float* __restrict__ Wgen,
    const float* __restrict__ bgen, float* __restrict__ sf)
{
  __shared__ __align__(16) float sSt[2][32][GEN_KC + 4];   // row stride 144B
  __shared__ __align__(16) float sW[2][128][GEN_KC + 4];   // row stride 144B
  int v0 = blockIdx.x * 128;
  int lane = threadIdx.x & 31, wave = threadIdx.x >> 5;
  int half = lane >> 4, lid = lane & 15;
  int rt = wave & 1;
  int cg = wave >> 1;            // 0..3, 32 cols each
  v8f acc0 = {}, acc1 = {};

  // prefetch chunk 0 into buffer 0
  {
    int r = threadIdx.x >> 3, kq = threadIdx.x & 7;
    ld_async16(&sSt[0][r][kq*4], st + (size_t)r*SS + kq*4);
    int unit = threadIdx.x;
    #pragma unroll
    for (int u = 0; u < 4; ++u) {
      int n = unit >> 3, kq2 = unit & 7;
      ld_async16(&sW[0][n][kq2*4], Wgen + ((size_t)(v0 + n))*SS + kq2*4);
      unit += 256;
    }
  }

  int buf = 0;
  for (int kc = 0; kc < SS; kc += GEN_KC) {
    int kn = kc + GEN_KC;
    if (kn < SS) {
      int nb = buf ^ 1;
      int r = threadIdx.x >> 3, kq = threadIdx.x & 7;
      ld_async16(&sSt[nb][r][kq*4], st + (size_t)r*SS + kn + kq*4);
      int unit = threadIdx.x;
      #pragma unroll
      for (int u = 0; u < 4; ++u) {
        int n = unit >> 3, kq2 = unit & 7;
        ld_async16(&sW[nb][n][kq2*4], Wgen + ((size_t)(v0 + n))*SS + kn + kq2*4);
        unit += 256;
      }
      async_wait5();       // current chunk complete; next-chunk ops in flight
    } else {
      async_wait0();
    }
    __syncthreads();
    #pragma unroll
    for (int kk = 0; kk < GEN_KC; kk += 4) {
      v2f a;
      a.x = sSt[buf][rt*16 + lid][kk + 2*half];
      a.y = sSt[buf][rt*16 + lid][kk + 2*half + 1];
      int c0 = cg*32 + lid;
      v2f b0, b1;
      b0.x = sW[buf][c0][kk + 2*half];       b0.y = sW[buf][c0][kk + 2*half + 1];
      b1.x = sW[buf][c0 + 16][kk + 2*half];  b1.y = sW[buf][c0 + 16][kk + 2*half + 1];
      acc0 = __builtin_amdgcn_wmma_f32_16x16x4_f32(false, a, false, b0,
               (short)0, acc0, false, false);
      acc1 = __builtin_amdgcn_wmma_f32_16x16x4_f32(false, a, false, b1,
               (short)0, acc1, false, false);
    }
    __syncthreads();       // all waves done reading buf before it is refilled
    buf ^= 1;
  }
  int vA = v0 + cg*32 + lid;
  int vB = vA + 16;
  float bg0 = bgen[vA], bg1 = bgen[vB];
  #pragma unroll
  for (int r = 0; r < 8; ++r) {
    int m = rt*16 + r + 8*half;
    sf[(size_t)m*NCLS + vA] = acc0[r] + bg0;
    sf[(size_t)m*NCLS + vB] = acc1[r] + bg1;
  }
}

// ---- K6: score_c[b,t] = sum_s tanh(X[b,t,:].Wc[s,:] + bc[s]) * st[b,s] -----
#define CP_KC 8
__global__ __launch_bounds__(256) void k_copyscore(
    const float* __restrict__ X, int Tn,
    const float* __restrict__ Wc, const float* __restrict__ bc,
    const float* __restrict__ st, float* __restrict__ outp)
{
  __shared__ __align__(16) float sA[16][CP_KC + 4];      // row stride 48B
  __shared__ __align__(16) float sW[SS][CP_KC + 4];      // row stride 48B
  __shared__ float sScore[16];
  int nt = Tn / 16;
  int b  = blockIdx.x / nt;
  int t0 = (blockIdx.x % nt) * 16;
  int lane = threadIdx.x & 31, wave = threadIdx.x >> 5;
  int half = lane >> 4, lid = lane & 15;
  int n0w = wave * 128;
  if (threadIdx.x < 16) sScore[threadIdx.x] = 0.0f;

  v8f zacc = {};
  v8f acc[8] = {zacc, zacc, zacc, zacc, zacc, zacc, zacc, zacc};

  for (int kc = 0; kc < HH; kc += CP_KC) {
    __syncthreads();
    // sA: 16 rows x 2 float4 = 32 units
    if (threadIdx.x < 32) {
      int r = threadIdx.x >> 1, kq = threadIdx.x & 1;
      ld_async16(&sA[r][kq*4],
                 X + ((size_t)b*Tn + t0 + r)*HH + kc + kq*4);
    }
    // sW: 1024 rows x 2 float4 = 2048 units, 8 per thread
    {
      int unit = threadIdx.x;
      #pragma unroll
      for (int u = 0; u < 8; ++u) {
        int s = unit >> 1, kq = unit & 1;
        ld_async16(&sW[s][kq*4], Wc + (size_t)s*HH + kc + kq*4);
        unit += 256;
      }
    }
    async_wait0();
    __syncthreads();
    #pragma unroll
    for (int kk = 0; kk < CP_KC; kk += 4) {
      v2f a;
      a.x = sA[lid][kk + 2*half];
      a.y = sA[lid][kk + 2*half + 1];
      #pragma unroll
      for (int ct = 0; ct < 8; ++ct) {
        int n = n0w + ct*16 + lid;
        v2f bf;
        bf.x = sW[n][kk + 2*half];
        bf.y = sW[n][kk + 2*half + 1];
        acc[ct] = __builtin_amdgcn_wmma_f32_16x16x4_f32(false, a, false, bf,
                    (short)0, acc[ct], false, false);
      }
    }
  }

  float p[8];
  #pragma unroll
  for (int r = 0; r < 8; ++r) p[r] = 0.0f;
  #pragma unroll
  for (int ct = 0; ct < 8; ++ct) {
    int n = n0w + ct*16 + lid;
    float bcv = bc[n];
    float stv = st[b*SS + n];
    #pragma unroll
    for (int r = 0; r < 8; ++r)
      p[r] += tanhf(acc[ct][r] + bcv) * stv;
  }
  #pragma unroll
  for (int r = 0; r < 8; ++r)
    for (int off = 8; off >= 1; off >>= 1)
      p[r] += __shfl_xor(p[r], off, 32);
  if (lid == 0) {
    #pragma unroll
    for (int r = 0; r < 8; ++r)
      atomicAdd(&sScore[r + 8*half], p[r]);
  }
  __syncthreads();
  if (threadIdx.x < 16)
    outp[(size_t)b*NCLS + t0 + threadIdx.x] = sScore[threadIdx.x];
}

// --------------- K7: per-batch max & sum(exp) over 32192 logits -------------
__global__ __launch_bounds__(256) void k_maxz(const float* __restrict__ sf,
                                              float* __restrict__ mz)
{
  __shared__ float red[256];
  int b = blockIdx.x, tid = threadIdx.x;
  const float* row = sf + (size_t)b*NCLS;
  float mx = -3.0e38f;
  for (int i = tid; i < NCLS; i += 256) mx = fmaxf(mx, row[i]);
  red[tid] = mx; __syncthreads();
  for (int s = 128; s > 0; s >>= 1) {
    if (tid < s) red[tid] = fmaxf(red[tid], red[tid + s]);
    __syncthreads();
  }
  mx = red[0]; __syncthreads();
  float sum = 0.0f;
  for (int i = tid; i < NCLS; i += 256) sum += expf(row[i] - mx);
  red[tid] = sum; __syncthreads();
  for (int s = 128; s > 0; s >>= 1) {
    if (tid < s) red[tid] += red[tid + s];
    __syncthreads();
  }
  if (tid == 0) { mz[b*2] = mx; mz[b*2+1] = red[0]; }
}

// ---- K8: out[b,v] = prob_g + sum_t pc_pz[t]*pv_z_prob[b,t,v] (dense) -------
__global__ __launch_bounds__(256) void k_out_dense(
    const float* __restrict__ sf, const float* __restrict__ mz,
    const float* __restrict__ pzprob, float* __restrict__ out)
{
  __shared__ float spz[TZZ];
  int b = blockIdx.y;
  int v = blockIdx.x*256 + threadIdx.x;
  float mx = mz[b*2], invZ = 1.0f / mz[b*2+1];
  if (threadIdx.x < TZZ)
    spz[threadIdx.x] = expf(sf[(size_t)b*NCLS + VV + TUU + threadIdx.x] - mx) * invZ;
  __syncthreads();
  float accv = expf(sf[(size_t)b*NCLS + v] - mx) * invZ;
  const float* pz = pzprob + (size_t)b*TZZ*VV + v;
  #pragma unroll 4
  for (int t = 0; t < TZZ; ++t) {
    if (t + 8 < TZZ) __builtin_prefetch(pz + (size_t)(t + 8)*VV, 0, 1);
    accv += spz[t] * pz[(size_t)t*VV];
  }
  out[(size_t)b*VV + v] = accv;
}

// ---- K9: scatter of copy-u probs via indices (replaces 524MB one-hot) ------
__global__ __launch_bounds__(256) void k_scatter(
    const float* __restrict__ sf, const float* __restrict__ mz,
    const int* __restrict__ uin, float* __restrict__ out)
{
  int idx = blockIdx.x*256 + threadIdx.x;        // 0..4095
  int b = idx / TUU, t = idx % TUU;
  float mx = mz[b*2], invZ = 1.0f / mz[b*2+1];
  float p = expf(sf[(size_t)b*NCLS + VV + t] - mx) * invZ;
  atomicAdd(&out[(size_t)b*VV + uin[idx]], p);
}

extern "C" void kernel_launch(void* const* d_in, const int* in_sizes, int n_in,
                              void* d_out, int out_size, void* d_ws, size_t ws_size,
                              hipStream_t stream)
{
  const int*   u_input = (const int*)  d_in[0];
  // d_in[1] = u_input_1hot: intentionally unused (index scatter instead)
  const float* uh      = (const float*)d_in[2];
  const float* pzprob  = (const float*)d_in[3];
  const float* pzh     = (const float*)d_in[4];
  const float* emb     = (const float*)d_in[5];
  const float* lasth   = (const float*)d_in[6];
  const float* Wattn   = (const float*)d_in[7];
  const float* battn   = (const float*)d_in[8];
  const float* vattn   = (const float*)d_in[9];
  const float* Wih     = (const float*)d_in[10];
  const float* Whh     = (const float*)d_in[11];
  const float* bih     = (const float*)d_in[12];
  const float* bhh     = (const float*)d_in[13];
  const float* Wgen    = (const float*)d_in[14];
  const float* bgen    = (const float*)d_in[15];
  const float* Wcpu    = (const float*)d_in[16];
  const float* bcpu    = (const float*)d_in[17];
  const float* Wcppz   = (const float*)d_in[18];
  const float* bcppz   = (const float*)d_in[19];

  float* ws     = (float*)d_ws;
  float* hW1    = ws + WS_HW1;
  float* scores = ws + WS_SCORES;
  float* x      = ws + WS_X;
  float* hnew   = ws + WS_HNEW;
  float* st     = ws + WS_ST;
  float* sf     = ws + WS_SF;
  float* mz     = ws + WS_MZ;

  float* out     = (float*)d_out;
  float* out_nlh = out + (size_t)BB*VV;
  float* out_gru = out_nlh + (size_t)BB*HH;

  k_hw1<<<BB, 256, 0, stream>>>(Wattn, battn, lasth, hW1);
  k_attn_scores<<<BB*(TT/16), 256, 0, stream>>>(uh, pzh, Wattn, hW1, vattn, scores);
  k_softmax_ctx<<<BB, 256, 0, stream>>>(uh, pzh, emb, scores, x);
  k_gru<<<BB, 256, 0, stream>>>(Wih, Whh, bih, bhh, x, lasth, emb,
                                st, hnew, out_nlh, out_gru);
  k_gen<<<VV/128, 256, 0, stream>>>(st, Wgen, bgen, sf);
  k_copyscore<<<BB*(TUU/16), 256, 0, stream>>>(uh,  TUU, Wcpu,  bcpu,  st, sf + VV);
  k_copyscore<<<BB*(TZZ/16), 256, 0, stream>>>(pzh, TZZ, Wcppz, bcppz, st, sf + VV + TUU);
  k_maxz<<<BB, 256, 0, stream>>>(sf, mz);
  dim3 g8(VV/256, BB);
  k_out_dense<<<g8, 256, 0, stream>>>(sf, mz, pzprob, out);
  k_scatter<<<(BB*TUU)/256, 256, 0, stream>>>(sf, mz, u_input, out);
}